// DepthConditionModel_68762426409363
// MI455X (gfx1250) — compile-verified
//
#include <hip/hip_runtime.h>
#include <math.h>

// ---------------- problem constants (from reference) ----------------
#define IMG_H 480
#define IMG_W 640
#define IMG_PIXELS (IMG_H * IMG_W)      // 307200
#define IMG_F4 (IMG_PIXELS / 4)         // 76800 float4 per image (640 % 4 == 0, no row wrap)
#define WX 256
#define WY 256
#define WXY (WX * WY)                   // 65536 bins

// ---------------- tuning ----------------
#define TPB 1024                        // 32 wave32s -> fills one WGP (256KB LDS => 1 WG/WGP)
#define NSPLIT 8                        // workgroups per image -> 128 WGs total
#define CHUNK_F4 (IMG_F4 / NSPLIT)      // 9600 float4 per WG
#define NTILES ((CHUNK_F4 + TPB - 1) / TPB)   // 10 tiles (last partial: 384)
#define VEC_PER_THREAD (WXY / 4 / TPB)  // 16 uint4 per thread over the bin map

// ---------------- CDNA5 async-to-LDS path (guarded; falls back cleanly) ----------
#if defined(__has_builtin)
#if __has_builtin(__builtin_amdgcn_global_load_async_to_lds_b128) && \
    __has_builtin(__builtin_amdgcn_s_wait_asynccnt)
#define HAVE_ASYNC_LDS 1
#endif
#endif
#ifndef HAVE_ASYNC_LDS
#define HAVE_ASYNC_LDS 0
#endif

#if HAVE_ASYNC_LDS
typedef int v4i_t __attribute__((ext_vector_type(4)));
typedef __attribute__((address_space(1))) v4i_t gv4i_t;   // global (device) int4
typedef __attribute__((address_space(3))) v4i_t lv4i_t;   // LDS int4

// generic -> addrspace(1): identical 64-bit address
__device__ static inline gv4i_t* to_glb4(const void* p) {
  return (gv4i_t*)(unsigned long long)p;
}
// generic -> addrspace(3): low 32 bits of the flat LDS-aperture address are the LDS offset
__device__ static inline lv4i_t* to_lds4(void* p) {
  return (lv4i_t*)(unsigned)(unsigned long long)p;
}
#endif

// ---------------- kernel 0: zero u32 accumulators ----------------
__global__ void k_zero(uint4* __restrict__ p, int n4) {
  int i = blockIdx.x * blockDim.x + threadIdx.x;
  if (i < n4) p[i] = make_uint4(0u, 0u, 0u, 0u);
}

// ---------------- kernel 1: per-image BEV count histogram ----------------
// grid = (NSPLIT, B); dynamic LDS = 256KB histogram + 32KB double-buffer stage
__global__ __launch_bounds__(TPB) void k_hist(const float* __restrict__ depth,
                                              unsigned* __restrict__ accum) {
  extern __shared__ unsigned smem[];
  unsigned* hist = smem;                          // WXY u32 = 256 KB
  float4* stage = (float4*)(smem + WXY);          // 2 * TPB float4 = 32 KB

  const int tid   = threadIdx.x;
  const int split = blockIdx.x;
  const int b     = blockIdx.y;

  // zero LDS histogram (ds_store_b128)
  uint4* h4 = (uint4*)hist;
#pragma unroll
  for (int k = 0; k < VEC_PER_THREAD; ++k)
    h4[tid + k * TPB] = make_uint4(0u, 0u, 0u, 0u);
  __syncthreads();

  const float4* img = (const float4*)(depth + (size_t)b * IMG_PIXELS);
  const int base = split * CHUNK_F4;

  auto process = [&](int q /* float4 index within image */, float4 dv) {
    int p  = q << 2;
    int v  = p / IMG_W;
    int u0 = p - v * IMG_W;
    float vf = (float)v - 240.0f;                 // v - CY
    float d4[4] = {dv.x, dv.y, dv.z, dv.w};
#pragma unroll
    for (int e = 0; e < 4; ++e) {
      float d = d4[e];
      float x = (((float)(u0 + e)) - 320.0f) * d / 1000.0f;   // (u-cx)*d/fx
      float y = vf * d / 850.0f;                              // (v-cy)*d/fy
      bool m = (x >= -51.2f) & (x <= 51.2f) & (y >= -51.2f) & (y <= 51.2f);
      int i = (int)floorf((x + 51.2f) / 0.4f);
      int j = (int)floorf((y + 51.2f) / 0.4f);
      i = min(max(i, 0), WX - 1);
      j = min(max(j, 0), WY - 1);
      if (m) atomicAdd(&hist[(i << 8) + j], 1u);              // ds_add_u32
    }
  };

#if HAVE_ASYNC_LDS
  // ---- double-buffered async global->LDS staging of the depth stream ----
  // prologue: stage tile 0 (always full)
  __builtin_amdgcn_global_load_async_to_lds_b128(
      to_glb4(&img[base + tid]), to_lds4(&stage[tid]), 0, 0);
  for (int t = 0; t + 1 < NTILES; ++t) {
    int nrel = (t + 1) * TPB + tid;
    if (nrel < CHUNK_F4)
      __builtin_amdgcn_global_load_async_to_lds_b128(
          to_glb4(&img[base + nrel]), to_lds4(&stage[((t + 1) & 1) * TPB + tid]), 0, 0);
    __builtin_amdgcn_s_wait_asynccnt(1);          // tile t resident (in-order completion)
    float4 dv = stage[(t & 1) * TPB + tid];       // ds_load_b128 (own slot only)
    process(base + t * TPB + tid, dv);
  }
  __builtin_amdgcn_s_wait_asynccnt(0);
  {
    int t = NTILES - 1;
    int rel = t * TPB + tid;
    if (rel < CHUNK_F4)
      process(base + rel, stage[(t & 1) * TPB + tid]);
  }
#else
  // fallback: direct b128 global loads
  for (int t = 0; t < NTILES; ++t) {
    int rel = t * TPB + tid;
    if (rel < CHUNK_F4) process(base + rel, img[base + rel]);
  }
#endif

  // ---- flush private histogram (b128 LDS reads); skip empty bins ----
  __syncthreads();
  unsigned* gout = accum + (size_t)b * WXY;
#pragma unroll
  for (int k = 0; k < VEC_PER_THREAD; ++k) {
    int m = tid + k * TPB;                        // uint4 index -> bins 4m..4m+3
    uint4 c = h4[m];                              // ds_load_b128
    int idx = m << 2;
    if (c.x) atomicAdd(&gout[idx + 0], c.x);      // global_atomic_add_u32
    if (c.y) atomicAdd(&gout[idx + 1], c.y);
    if (c.z) atomicAdd(&gout[idx + 2], c.z);
    if (c.w) atomicAdd(&gout[idx + 3], c.w);
  }
}

// ---------------- kernel 2: exact stats + in-place normalize ----------------
// grid = B blocks, TPB threads; each thread owns 64 *contiguous* bins
// (16 uint4 loads / 16 float4 stores; counts are integers -> int64 stats exact)
__global__ __launch_bounds__(TPB) void k_norm(float* __restrict__ out) {
  const int b = blockIdx.x, tid = threadIdx.x;
  const uint4* c4 = (const uint4*)((const unsigned*)out + (size_t)b * WXY) +
                    (size_t)tid * VEC_PER_THREAD;

  long long s = 0, q = 0;
#pragma unroll
  for (int k = 0; k < VEC_PER_THREAD; ++k) {
    uint4 c = c4[k];                              // global_load_b128
    long long x0 = c.x, x1 = c.y, x2 = c.z, x3 = c.w;
    s += x0 + x1 + x2 + x3;
    q += x0 * x0 + x1 * x1 + x2 * x2 + x3 * x3;
  }
  // wave32 butterfly reduction
#pragma unroll
  for (int off = 16; off; off >>= 1) {
    s += __shfl_xor(s, off, 32);
    q += __shfl_xor(q, off, 32);
  }
  __shared__ long long sS[32], sQ[32];
  __shared__ double sMean, sInv;
  int wid = tid >> 5, lane = tid & 31;
  if (lane == 0) { sS[wid] = s; sQ[wid] = q; }
  __syncthreads();
  if (wid == 0) {
    s = sS[lane];
    q = sQ[lane];
#pragma unroll
    for (int off = 16; off; off >>= 1) {
      s += __shfl_xor(s, off, 32);
      q += __shfl_xor(q, off, 32);
    }
    if (lane == 0) {
      double n    = (double)WXY;
      double mean = (double)s / n;
      double var  = ((double)q - (double)s * (double)s / n) / (n - 1.0);  // ddof=1
      sMean = mean;
      sInv  = 1.0 / sqrt(var);
    }
  }
  __syncthreads();
  double mean = sMean, inv = sInv;
  float4* o4 = (float4*)(out + (size_t)b * WXY) + (size_t)tid * VEC_PER_THREAD;
#pragma unroll
  for (int k = 0; k < VEC_PER_THREAD; ++k) {
    uint4 c = c4[k];                              // same addresses this thread owns
    float4 r;
    r.x = (float)(((double)c.x - mean) * inv);
    r.y = (float)(((double)c.y - mean) * inv);
    r.z = (float)(((double)c.z - mean) * inv);
    r.w = (float)(((double)c.w - mean) * inv);
    o4[k] = r;                                    // global_store_b128
  }
}

// ---------------- launcher ----------------
extern "C" void kernel_launch(void* const* d_in, const int* in_sizes, int n_in,
                              void* d_out, int out_size, void* d_ws, size_t ws_size,
                              hipStream_t stream) {
  (void)n_in; (void)out_size; (void)d_ws; (void)ws_size;
  const float* depth = (const float*)d_in[0];
  const int B = in_sizes[0] / IMG_PIXELS;         // 16

  // zero u32 accumulators living in d_out
  int n4 = (B * WXY) / 4;
  hipLaunchKernelGGL(k_zero, dim3((n4 + 255) / 256), dim3(256), 0, stream,
                     (uint4*)d_out, n4);

  // histogram: 256KB hist + 32KB async stage
  size_t lds_bytes = (size_t)WXY * sizeof(unsigned) + 2u * TPB * sizeof(float4);
  hipLaunchKernelGGL(k_hist, dim3(NSPLIT, B), dim3(TPB), lds_bytes, stream,
                     depth, (unsigned*)d_out);

  // stats + normalize in place
  hipLaunchKernelGGL(k_norm, dim3(B), dim3(TPB), 0, stream, (float*)d_out);
}